// PyramidRoiAlignLayer_77627238908019
// MI455X (gfx1250) — compile-verified
//
#include <hip/hip_runtime.h>
#include <math.h>

// ---------------------------------------------------------------------------
// Kernel 1: level compute + stable counting sort of <=1024 ROIs by level (0..3)
// Packed 4x16-bit counts scanned with Hillis-Steele in LDS.
// ---------------------------------------------------------------------------
__global__ __launch_bounds__(1024)
void roi_level_sort_kernel(const float* __restrict__ rois, int N,
                           int* __restrict__ order, int* __restrict__ lvls)
{
    __shared__ unsigned long long scan[1024];
    const int i = threadIdx.x;

    int k = 0;
    unsigned long long v = 0ull;
    if (i < N) {
        const float y1 = rois[i * 4 + 0];
        const float x1 = rois[i * 4 + 1];
        const float y2 = rois[i * 4 + 2];
        const float x2 = rois[i * 4 + 3];
        const float h = y2 - y1;
        const float w = x2 - x1;
        // lvl = clip(round(log2(sqrt(h*w)) - 5), 0, 3); jnp.round == rint (half-even)
        const float l = 0.5f * log2f(h * w) - 5.0f;
        int ki = (int)rintf(l);
        k = ki < 0 ? 0 : (ki > 3 ? 3 : ki);
        v = 1ull << (16 * k);
    }
    scan[i] = v;
    __syncthreads();

    #pragma unroll
    for (int off = 1; off < 1024; off <<= 1) {
        const unsigned long long add = (i >= off) ? scan[i - off] : 0ull;
        __syncthreads();
        scan[i] += add;
        __syncthreads();
    }

    if (i < N) {
        const unsigned long long total = scan[1023];
        const unsigned long long excl  = scan[i] - v;            // exclusive prefix
        const int rank = (int)((excl >> (16 * k)) & 0xFFFFull);  // stable rank in level
        int base = 0;
        for (int l = 0; l < k; ++l) base += (int)((total >> (16 * l)) & 0xFFFFull);
        order[base + rank] = i;
        lvls[base + rank]  = k;
    }
}

// ---------------------------------------------------------------------------
// CDNA5 async global->LDS copy: one B128 per lane (16B), one ASYNCcnt per wave.
// ---------------------------------------------------------------------------
__device__ __forceinline__ void async_ld_b128(unsigned lds_byte_off, const float* gaddr)
{
    asm volatile("global_load_async_to_lds_b128 %0, %1, off"
                 :: "v"(lds_byte_off), "v"(gaddr)
                 : "memory");
}

#define WAIT_ASYNCCNT(n) asm volatile("s_wait_asynccnt %0" :: "i"(n) : "memory")

// ---------------------------------------------------------------------------
// Kernel 2: one block per (sorted slot, py). 256 threads = 1 channel each.
// 4 corner rows (4 x 1KB) staged into LDS by 8 wave-level async B128 ops,
// double-buffered across the 7 px positions.
// ---------------------------------------------------------------------------
__global__ __launch_bounds__(256)
void roi_gather_kernel(const float* __restrict__ f0, const float* __restrict__ f1,
                       const float* __restrict__ f2, const float* __restrict__ f3,
                       const float* __restrict__ rois,
                       const int* __restrict__ order, const int* __restrict__ lvls,
                       float* __restrict__ out)
{
    extern __shared__ float lds[];          // 2 buffers * 4 rows * 256 floats = 8KB

    const int py   = blockIdx.x;            // 0..6
    const int slot = blockIdx.y;            // sorted output slot
    const int tid  = threadIdx.x;           // channel in compute phase

    const int src = order[slot];
    const int lvl = lvls[slot];
    const float* feat = (lvl == 0) ? f0 : (lvl == 1) ? f1 : (lvl == 2) ? f2 : f3;
    const int   W    = 256 >> lvl;          // H == W per level
    const float Hm1  = (float)(W - 1);

    const float inv = 1.0f / 1024.0f;
    const float ry1 = rois[src * 4 + 0] * inv;
    const float rx1 = rois[src * 4 + 1] * inv;
    const float ry2 = rois[src * 4 + 2] * inv;
    const float rx2 = rois[src * 4 + 3] * inv;

    // y sample for this py (TF crop_and_resize grid)
    const float iy  = ry1 * Hm1 + (float)py * (((ry2 - ry1) * Hm1) / 6.0f);
    const bool  vy  = (iy >= 0.0f) && (iy <= Hm1);
    const float y0f = floorf(iy);
    const float ly  = iy - y0f;
    int y0  = (int)y0f;     y0  = y0  < 0 ? 0 : (y0  > W - 1 ? W - 1 : y0);
    int y1i = (int)y0f + 1; y1i = y1i < 0 ? 0 : (y1i > W - 1 ? W - 1 : y1i);

    // all 7 x samples up front (kept in registers; loop fully unrolled below)
    float lx[7]; int xs0[7], xs1[7]; bool vx[7];
    const float xstep = ((rx2 - rx1) * Hm1) / 6.0f;
    #pragma unroll
    for (int px = 0; px < 7; ++px) {
        const float ix  = rx1 * Hm1 + (float)px * xstep;
        vx[px] = (ix >= 0.0f) && (ix <= Hm1);
        const float x0f = floorf(ix);
        lx[px] = ix - x0f;
        int x0 = (int)x0f;     xs0[px] = x0 < 0 ? 0 : (x0 > W - 1 ? W - 1 : x0);
        int x1 = (int)x0f + 1; xs1[px] = x1 < 0 ? 0 : (x1 > W - 1 ? W - 1 : x1);
    }

    // Async staging layout: thread t -> corner = t>>6 in {(y0,x0),(y0,x1),(y1,x0),(y1,x1)},
    // 4 consecutive channels per lane (B128). 8 waves move exactly the 4KB stage.
    const int corner = tid >> 6;
    const int elem   = (tid & 63) << 2;
    const int cy     = (corner & 2) ? y1i : y0;

    auto issue = [&](int px, int buf) {
        const int cx = (corner & 1) ? xs1[px] : xs0[px];
        const float* g = feat + ((size_t)cy * W + cx) * 256 + elem;
        const unsigned loff = (unsigned)((buf * 1024 + corner * 256 + elem) * 4);
        async_ld_b128(loff, g);
    };

    issue(0, 0);
    #pragma unroll
    for (int px = 0; px < 7; ++px) {
        const int buf = px & 1;
        if (px < 6) {
            issue(px + 1, buf ^ 1);
            WAIT_ASYNCCNT(1);       // stage px landed (per-wave in-order completion)
        } else {
            WAIT_ASYNCCNT(0);
        }
        __syncthreads();            // all 8 waves' quarters visible in LDS

        const float* b = lds + buf * 1024;
        const float f00 = b[0 * 256 + tid];
        const float f01 = b[1 * 256 + tid];
        const float f10 = b[2 * 256 + tid];
        const float f11 = b[3 * 256 + tid];
        const float top = f00 + (f01 - f00) * lx[px];
        const float bot = f10 + (f11 - f10) * lx[px];
        float val = top + (bot - top) * ly;
        val = (vy && vx[px]) ? val : 0.0f;

        const size_t oidx = (((size_t)slot * 7 + py) * 7 + px) * 256 + tid;
        __builtin_nontemporal_store(val, &out[oidx]);   // keep pyramids L2-resident

        __syncthreads();            // buffer 'buf' is overwritten at stage px+2
    }
}

// ---------------------------------------------------------------------------
// Host launcher
// ---------------------------------------------------------------------------
extern "C" void kernel_launch(void* const* d_in, const int* in_sizes, int n_in,
                              void* d_out, int out_size, void* d_ws, size_t ws_size,
                              hipStream_t stream)
{
    const float* f0   = (const float*)d_in[0];
    const float* f1   = (const float*)d_in[1];
    const float* f2   = (const float*)d_in[2];
    const float* f3   = (const float*)d_in[3];
    const float* rois = (const float*)d_in[4];
    const int N = in_sizes[4] / 4;          // 1000 (<=1024 required by sorter)

    int* order = (int*)d_ws;
    int* lvls  = order + N;

    roi_level_sort_kernel<<<1, 1024, 0, stream>>>(rois, N, order, lvls);
    roi_gather_kernel<<<dim3(7, N), 256, 8192, stream>>>(f0, f1, f2, f3, rois,
                                                         order, lvls, (float*)d_out);
}